// ExpectedLikelihoodKernel_26680336843242
// MI455X (gfx1250) — compile-verified
//
#include <hip/hip_runtime.h>
#include <hip/hip_bf16.h>

typedef float v2f __attribute__((ext_vector_type(2)));
typedef float v8f __attribute__((ext_vector_type(8)));
typedef int   v4i __attribute__((ext_vector_type(4)));

#define B_ROWS 2048
#define C_COLS 16384
#define DIM 128
#define ROWS_PER_BLOCK 16
#define COLS_PER_ITER 128
#define NSPLIT 4
#define ITERS_PER_SPLIT (C_COLS / NSPLIT / COLS_PER_ITER) // 32
#define LDA 132
#define LDB 132
#define PARTS_STRIDE 33  // 16 x (M,S) + 1 x G per block

#if __has_builtin(__builtin_amdgcn_global_load_async_to_lds_b128) && \
    __has_builtin(__builtin_amdgcn_s_wait_asynccnt)
#define HAVE_ASYNC_LDS 1
#else
#define HAVE_ASYNC_LDS 0
#endif

// log C_128(kappa) via the uniform asymptotic expansion, algebraically
// simplified (exactly equal to the reference formula for dim=128):
//   v = 63, s = sqrt(v^2 + k^2)
//   logC = v*log(v+s) - s + 0.5*log(s) - 63.5*log(2*pi)
__device__ __forceinline__ float log_vmf_c(float kappa) {
    const float v = 63.0f;
    const float CST = -63.5f * 1.8378770664093453f; // -63.5 * log(2*pi)
    float s = sqrtf(fmaf(kappa, kappa, v * v));
    return fmaf(v, __logf(v + s), fmaf(0.5f, __logf(s), CST - s));
}

#if HAVE_ASYNC_LDS
// Cooperative async stage of one 128x128 f32 class tile into LDS.
// 16 x b128 async loads per thread (uniform per wave -> ASYNCcnt += 16/wave).
__device__ __forceinline__ void issue_tile_async(const float* __restrict__ src,
                                                 float* dstLds, int t) {
#pragma unroll
    for (int i = 0; i < 16; ++i) {
        int q = t + i * 256;            // 4096 float4 total
        int col = q >> 5, k4 = q & 31;
        __builtin_amdgcn_global_load_async_to_lds_b128(
            (v4i*)(src + (size_t)col * DIM + k4 * 4),
            (v4i*)(dstLds + col * LDB + k4 * 4), 0, 0);
    }
}
#endif

// One wave per class row: normalize, kappa2, logC(kappa2).
__global__ void class_stats_kernel(const float* __restrict__ W,
                                   float* __restrict__ cls_mu,
                                   float* __restrict__ kap2,
                                   float* __restrict__ lc2) {
    int row  = (blockIdx.x * blockDim.x + threadIdx.x) >> 5;
    int lane = threadIdx.x & 31;
    if (row >= C_COLS) return;
    float4 w4 = *(const float4*)(W + (size_t)row * DIM + lane * 4);
    float ssq = w4.x * w4.x + w4.y * w4.y + w4.z * w4.z + w4.w * w4.w;
    for (int off = 16; off > 0; off >>= 1) ssq += __shfl_xor(ssq, off, 32);
    float nrm = sqrtf(ssq);
    float inv = 1.0f / fmaxf(nrm, 1e-12f);
    float4 mu; mu.x = w4.x * inv; mu.y = w4.y * inv; mu.z = w4.z * inv; mu.w = w4.w * inv;
    *(float4*)(cls_mu + (size_t)row * DIM + lane * 4) = mu;
    if (lane == 0) {
        float kp = fmaxf(nrm, 1.0f) * 10.0f;
        kap2[row] = kp;
        lc2[row]  = log_vmf_c(kp);
    }
}

// One wave per batch row: normalize, kappa1 = 1/unc, logC(kappa1).
__global__ void batch_stats_kernel(const float* __restrict__ F,
                                   const float* __restrict__ unc,
                                   float* __restrict__ bat_mu,
                                   float* __restrict__ k1a,
                                   float* __restrict__ lc1a) {
    int row  = (blockIdx.x * blockDim.x + threadIdx.x) >> 5;
    int lane = threadIdx.x & 31;
    if (row >= B_ROWS) return;
    float4 f4 = *(const float4*)(F + (size_t)row * DIM + lane * 4);
    float ssq = f4.x * f4.x + f4.y * f4.y + f4.z * f4.z + f4.w * f4.w;
    for (int off = 16; off > 0; off >>= 1) ssq += __shfl_xor(ssq, off, 32);
    float nrm = sqrtf(ssq);
    float inv = 1.0f / fmaxf(nrm, 1e-12f);
    float4 mu; mu.x = f4.x * inv; mu.y = f4.y * inv; mu.z = f4.z * inv; mu.w = f4.w * inv;
    *(float4*)(bat_mu + (size_t)row * DIM + lane * 4) = mu;
    if (lane == 0) {
        float kp = 1.0f / unc[row];
        k1a[row]  = kp;
        lc1a[row] = log_vmf_c(kp);
    }
}

// Main fused kernel: 16-row strip x 1/NSPLIT of the columns per block.
// 8 waves, each owns a 16x16 WMMA tile (fp32 WMMA, K=128 -> 32 x 16x16x4).
// Class tiles stream through LDS, double-buffered via async global->LDS.
__launch_bounds__(256, 1)
__global__ void vmf_main_kernel(const float* __restrict__ cls_mu,
                                const float* __restrict__ bat_mu,
                                const float* __restrict__ kap2,
                                const float* __restrict__ lc2,
                                const float* __restrict__ k1a,
                                const float* __restrict__ lc1a,
                                const int* __restrict__ y,
                                float* __restrict__ parts) {
    __shared__ __align__(16) float Asm[ROWS_PER_BLOCK * LDA];     //  8.4 KB
#if HAVE_ASYNC_LDS
    __shared__ __align__(16) float Bsm[2][COLS_PER_ITER * LDB];   // 135 KB
#else
    __shared__ __align__(16) float Bsm[1][COLS_PER_ITER * LDB];   // 67.6 KB
#endif
    __shared__ float K1s[ROWS_PER_BLOCK], L1s[ROWS_PER_BLOCK];
    __shared__ int   Ys[ROWS_PER_BLOCK];
    __shared__ float redM[ROWS_PER_BLOCK][8], redS[ROWS_PER_BLOCK][8];
    __shared__ float GWs[8];

    const int t     = threadIdx.x;
    const int w     = t >> 5;         // wave 0..7
    const int lane  = t & 31;
    const int mlane = lane & 15;
    const int hi    = lane >> 4;      // K/M half selector per ISA layout
    const int row0  = blockIdx.x * ROWS_PER_BLOCK;
    const int cn    = w * 16 + mlane; // column within the 128-wide tile

    // Stage A tile (batch_mu, 16x128) + per-row stats.
#pragma unroll
    for (int i = 0; i < 2; ++i) {
        int q = t + i * 256;            // 512 float4 total
        int m = q >> 5, k4 = q & 31;
        float4 v = *(const float4*)(bat_mu + (size_t)(row0 + m) * DIM + k4 * 4);
        *(float4*)(&Asm[m * LDA + k4 * 4]) = v;
    }
    if (t < ROWS_PER_BLOCK) {
        K1s[t] = k1a[row0 + t];
        L1s[t] = lc1a[row0 + t];
        Ys[t]  = y[row0 + t];
    }

    float m8[8], s8[8];
    float gsum = 0.0f;
#pragma unroll
    for (int r = 0; r < 8; ++r) { m8[r] = -3.0e38f; s8[r] = 0.0f; }

#if HAVE_ASYNC_LDS
    // Prologue: async-stage tile 0 into buffer 0.
    issue_tile_async(cls_mu + (size_t)(blockIdx.y * ITERS_PER_SPLIT) * COLS_PER_ITER * DIM,
                     &Bsm[0][0], t);
#endif

    for (int it = 0; it < ITERS_PER_SPLIT; ++it) {
        int n0 = (blockIdx.y * ITERS_PER_SPLIT + it) * COLS_PER_ITER;
        const float* Bbuf;
#if HAVE_ASYNC_LDS
        int cur = it & 1;
        if (it + 1 < ITERS_PER_SPLIT) {
            // Overlap: issue next tile into the other buffer, then wait only
            // for the older 16 per-wave async loads (in-order completion).
            issue_tile_async(cls_mu + ((size_t)n0 + COLS_PER_ITER) * DIM,
                             &Bsm[cur ^ 1][0], t);
            __builtin_amdgcn_s_wait_asynccnt(16);
        } else {
            __builtin_amdgcn_s_wait_asynccnt(0);
        }
        __syncthreads();                 // all waves' portions of tile `it` visible
        Bbuf = &Bsm[cur][0];
#else
        __syncthreads();
#pragma unroll
        for (int i = 0; i < 16; ++i) {
            int q = t + i * 256;         // 4096 float4 total
            int col = q >> 5, k4 = q & 31;
            float4 v = *(const float4*)(cls_mu + (size_t)(n0 + col) * DIM + k4 * 4);
            *(float4*)(&Bsm[0][col * LDB + k4 * 4]) = v;
        }
        if (it + 1 < ITERS_PER_SPLIT)    // warm next tile (global_prefetch_b8)
            __builtin_prefetch(cls_mu + (size_t)(n0 + COLS_PER_ITER) * DIM + t * 64, 0, 1);
        __syncthreads();
        Bbuf = &Bsm[0][0];
#endif

        // Per-lane column stats straight from L2 (16 consecutive cols/wave).
        float k2v = __ldg(kap2 + n0 + cn);
        float l2v = __ldg(lc2 + n0 + cn);

        // 16x16 fp32 tile, K=128 via 32 x v_wmma_f32_16x16x4_f32.
        v8f acc = {};
#pragma unroll
        for (int kk = 0; kk < 32; ++kk) {
            v2f a = *(const v2f*)(&Asm [mlane * LDA + kk * 4 + 2 * hi]);
            v2f b = *(const v2f*)(&Bbuf[cn    * LDB + kk * 4 + 2 * hi]);
            acc = __builtin_amdgcn_wmma_f32_16x16x4_f32(
                false, a, false, b, (short)0, acc, false, false);
        }

        // Fused epilogue: sim + online logsumexp + y-gather.
        int gcol = n0 + cn;
#pragma unroll
        for (int r = 0; r < 8; ++r) {
            int mrow = r + 8 * hi;       // C/D layout: VGPR r -> row r + 8*hi
            float k1v = K1s[mrow], l1v = L1s[mrow];
            float d   = acc[r];
            float t2  = fmaf(2.0f * k1v * k2v, d, fmaf(k1v, k1v, k2v * k2v));
            float kp3 = sqrtf(fmaxf(t2, 1e-12f));
            float sim = l1v + l2v - log_vmf_c(kp3);
            float mo  = m8[r];
            float mn  = fmaxf(mo, sim);
            s8[r] = fmaf(s8[r], __expf(mo - mn), __expf(sim - mn));
            m8[r] = mn;
            if (gcol == Ys[mrow]) gsum += sim;
        }
#if HAVE_ASYNC_LDS
        __syncthreads();                 // done reading buf `cur` before reuse
#endif
    }

    // In-wave merge of (max,sum) partials across the 16-lane half (rows match
    // the C/D layout), then tiny LDS combine across the 8 waves.
#pragma unroll
    for (int r = 0; r < 8; ++r) {
        float M = m8[r], S = s8[r];
#pragma unroll
        for (int off = 8; off >= 1; off >>= 1) {
            float Mo = __shfl_xor(M, off, 32);
            float So = __shfl_xor(S, off, 32);
            float nm = fmaxf(M, Mo);
            S = fmaf(S, __expf(M - nm), So * __expf(Mo - nm));
            M = nm;
        }
        if (mlane == 0) { redM[r + 8 * hi][w] = M; redS[r + 8 * hi][w] = S; }
    }
    for (int off = 16; off >= 1; off >>= 1) gsum += __shfl_xor(gsum, off, 32);
    if (lane == 0) GWs[w] = gsum;
    __syncthreads();

    float* o = parts + (size_t)(blockIdx.x * NSPLIT + blockIdx.y) * PARTS_STRIDE;
    if (t < ROWS_PER_BLOCK) {
        float M = -3.0e38f, S = 0.0f;
        for (int j = 0; j < 8; ++j) {
            float mj = redM[t][j], sj = redS[t][j];
            float nm = fmaxf(M, mj);
            S = fmaf(S, __expf(M - nm), sj * __expf(mj - nm));
            M = nm;
        }
        o[2 * t] = M; o[2 * t + 1] = S;
    }
    if (t == 0) {
        float G = 0.0f;
        for (int j = 0; j < 8; ++j) G += GWs[j];
        o[32] = G;
    }
}

// Merge the NSPLIT column-splits per row, then deterministic mean.
__global__ void vmf_finalize_kernel(const float* __restrict__ parts,
                                    float* __restrict__ out) {
    __shared__ float red[256];
    int t = threadIdx.x;
    float acc = 0.0f;
    for (int row = t; row < B_ROWS; row += 256) {
        int rb = row >> 4, r = row & 15;
        float M = -3.0e38f, S = 0.0f, G = 0.0f;
        for (int g = 0; g < NSPLIT; ++g) {
            const float* pp = parts + (size_t)(rb * NSPLIT + g) * PARTS_STRIDE;
            float mj = pp[2 * r], sj = pp[2 * r + 1];
            float nm = fmaxf(M, mj);
            S = fmaf(S, __expf(M - nm), sj * __expf(mj - nm));
            M = nm;
            if (r == 0) G += pp[32];     // block gather-sums counted once
        }
        acc += (M + __logf(S)) - G;      // lse - gathered
    }
    red[t] = acc;
    __syncthreads();
    for (int off = 128; off > 0; off >>= 1) {
        if (t < off) red[t] += red[t + off];
        __syncthreads();
    }
    if (t == 0) out[0] = red[0] / (float)B_ROWS;
}

extern "C" void kernel_launch(void* const* d_in, const int* in_sizes, int n_in,
                              void* d_out, int out_size, void* d_ws, size_t ws_size,
                              hipStream_t stream) {
    // inputs: 0=pred (UNUSED by reference), 1=unc, 2=y, 3=features, 4=classifier_weight
    const float* unc      = (const float*)d_in[1];
    const int*   y        = (const int*)  d_in[2];
    const float* features = (const float*)d_in[3];
    const float* W        = (const float*)d_in[4];

    float* ws     = (float*)d_ws;
    float* cls_mu = ws;                                   // C*D   = 2,097,152 f
    float* bat_mu = cls_mu + (size_t)C_COLS * DIM;        // B*D   =   262,144 f
    float* kap2   = bat_mu + (size_t)B_ROWS * DIM;        // C
    float* lc2    = kap2 + C_COLS;                        // C
    float* k1a    = lc2 + C_COLS;                         // B
    float* lc1a   = k1a + B_ROWS;                         // B
    float* parts  = lc1a + B_ROWS;                        // 512 * 33

    class_stats_kernel<<<C_COLS / 8, 256, 0, stream>>>(W, cls_mu, kap2, lc2);
    batch_stats_kernel<<<B_ROWS / 8, 256, 0, stream>>>(features, unc, bat_mu, k1a, lc1a);
    dim3 grid(B_ROWS / ROWS_PER_BLOCK, NSPLIT);
    vmf_main_kernel<<<grid, 256, 0, stream>>>(cls_mu, bat_mu, kap2, lc2, k1a, lc1a, y, parts);
    vmf_finalize_kernel<<<1, 256, 0, stream>>>(parts, (float*)d_out);
}